// GGNNWithLocalGlobal_28621662060642
// MI455X (gfx1250) — compile-verified
//
#include <hip/hip_runtime.h>
#include <math.h>

#define H 128
#define NGRAPHS 256
#define NCLASSES 10

typedef __attribute__((ext_vector_type(2))) float v2f;
typedef __attribute__((ext_vector_type(8))) float v8f;

// ---------------------------------------------------------------------------
// Native fp32 WMMA: D(16x16,f32) = A(16x4,f32) * B(4x16,f32) + C
// 8-arg form: (neg_a, A, neg_b, B, c_mod, C, reuse_a, reuse_b)
// ---------------------------------------------------------------------------
__device__ __forceinline__ v8f wmma_f32(v2f a, v2f b, v8f c) {
    return __builtin_amdgcn_wmma_f32_16x16x4_f32(false, a, false, b, (short)0, c,
                                                 false, false);
}

// A fragment: rows row0..row0+15 of row-major A (lda), K-chunk k0..k0+3.
// ISA layout: lanes 0-15 -> M=lane, VGPR0=K0,VGPR1=K1; lanes 16-31 -> K2,K3.
__device__ __forceinline__ v2f load_a_frag(const float* __restrict__ A, int lda,
                                           int row0, int k0, int lane) {
    int r = row0 + (lane & 15);
    int k = k0 + ((lane >> 4) << 1);
    const float* p = A + (size_t)r * lda + k;
    v2f o; o.x = p[0]; o.y = p[1];
    return o;
}

// B fragment, B row-major KxN (ldb=N): lanes 0-15 -> N=col0+lane, K0/K1.
__device__ __forceinline__ v2f load_b_frag(const float* __restrict__ B, int ldb,
                                           int k0, int col0, int lane) {
    int n = col0 + (lane & 15);
    int k = k0 + ((lane >> 4) << 1);
    v2f o; o.x = B[(size_t)k * ldb + n]; o.y = B[(size_t)(k + 1) * ldb + n];
    return o;
}

// B fragment from transposed storage Bt (N rows x K cols, row-major):
// B[k][n] = Bt[n][k]  -> contiguous float2 load.
__device__ __forceinline__ v2f load_bt_frag(const float* __restrict__ Bt, int ldbt,
                                            int k0, int col0, int lane) {
    int n = col0 + (lane & 15);
    int k = k0 + ((lane >> 4) << 1);
    const float* p = Bt + (size_t)n * ldbt + k;
    v2f o; o.x = p[0]; o.y = p[1];
    return o;
}

__device__ __forceinline__ float sigmoidf_(float x) {
    return 1.0f / (1.0f + expf(-x));
}

// ---------------------------------------------------------------------------
// Generic WMMA GEMM: C[M,N] = A[M,K] @ B (+bias) (+relu). One 16x16 tile/wave.
// ---------------------------------------------------------------------------
template <bool BT, bool BIAS, bool RELU>
__global__ __launch_bounds__(256) void gemm_kernel(
    const float* __restrict__ A, const float* __restrict__ B,
    const float* __restrict__ bias, float* __restrict__ C,
    int M, int N, int K) {
    const int lane = threadIdx.x & 31;
    const int wave = threadIdx.x >> 5;
    const int ntiles = N >> 4;
    const int mtiles = (M + 15) >> 4;
    const int tile = blockIdx.x * (blockDim.x >> 5) + wave;
    if (tile >= mtiles * ntiles) return;  // wave-uniform: EXEC stays all-ones
    const int row0 = (tile / ntiles) << 4;
    const int col0 = (tile % ntiles) << 4;

    v8f acc = {};
    for (int k0 = 0; k0 < K; k0 += 4) {
        v2f a = load_a_frag(A, K, row0, k0, lane);
        v2f b = BT ? load_bt_frag(B, K, k0, col0, lane)
                   : load_b_frag(B, N, k0, col0, lane);
        acc = wmma_f32(a, b, acc);
    }

    const int n = col0 + (lane & 15);
    const int mbase = row0 + ((lane >> 4) << 3);
    const float bv = BIAS ? bias[n] : 0.0f;
#pragma unroll
    for (int v = 0; v < 8; ++v) {
        float x = acc[v] + bv;
        if (RELU) x = fmaxf(x, 0.0f);
        C[(size_t)(mbase + v) * N + n] = x;
    }
}

// ---------------------------------------------------------------------------
// Fused GRU cell, in-place h update. One workgroup (8 waves) per 16-row tile.
// Wave j owns output columns [16j,16j+16); computes r/z/n gate tiles for both
// aggr@w_ih^T and h@w_hh^T with six WMMA accumulators; gate math in-register.
// ---------------------------------------------------------------------------
__global__ __launch_bounds__(256) void gru_kernel(
    const float* __restrict__ aggr, float* __restrict__ h,
    const float* __restrict__ w_ih, const float* __restrict__ w_hh,
    const float* __restrict__ b_ih, const float* __restrict__ b_hh) {
    const int lane = threadIdx.x & 31;
    const int j = threadIdx.x >> 5;      // column tile 0..7
    const int row0 = blockIdx.x << 4;
    const int cr = j * 16;               // r-gate columns in 3H space
    const int cz = H + j * 16;           // z-gate columns
    const int cn = 2 * H + j * 16;       // n-gate columns

    v8f ir = {}, iz = {}, in_ = {}, hr = {}, hz = {}, hn = {};
    for (int k0 = 0; k0 < H; k0 += 4) {
        v2f Aa = load_a_frag(aggr, H, row0, k0, lane);
        v2f Ah = load_a_frag(h, H, row0, k0, lane);
        v2f Bir = load_bt_frag(w_ih, H, k0, cr, lane);
        v2f Biz = load_bt_frag(w_ih, H, k0, cz, lane);
        v2f Bin = load_bt_frag(w_ih, H, k0, cn, lane);
        v2f Bhr = load_bt_frag(w_hh, H, k0, cr, lane);
        v2f Bhz = load_bt_frag(w_hh, H, k0, cz, lane);
        v2f Bhn = load_bt_frag(w_hh, H, k0, cn, lane);
        ir  = wmma_f32(Aa, Bir, ir);
        iz  = wmma_f32(Aa, Biz, iz);
        in_ = wmma_f32(Aa, Bin, in_);
        hr  = wmma_f32(Ah, Bhr, hr);
        hz  = wmma_f32(Ah, Bhz, hz);
        hn  = wmma_f32(Ah, Bhn, hn);
    }

    const int n16 = lane & 15;
    const int mbase = row0 + ((lane >> 4) << 3);
    // Load h tile in C-layout (before the barrier: all reads precede writes).
    float hc[8];
#pragma unroll
    for (int v = 0; v < 8; ++v)
        hc[v] = h[(size_t)(mbase + v) * H + j * 16 + n16];
    const float bir = b_ih[cr + n16], biz = b_ih[cz + n16], bin = b_ih[cn + n16];
    const float bhr = b_hh[cr + n16], bhz = b_hh[cz + n16], bhn = b_hh[cn + n16];

    __syncthreads();  // workgroup fence: every wave's h-reads done before writes

#pragma unroll
    for (int v = 0; v < 8; ++v) {
        float r = sigmoidf_((ir[v] + bir) + (hr[v] + bhr));
        float z = sigmoidf_((iz[v] + biz) + (hz[v] + bhz));
        float nn = tanhf((in_[v] + bin) + r * (hn[v] + bhn));
        h[(size_t)(mbase + v) * H + j * 16 + n16] = (1.0f - z) * nn + z * hc[v];
    }
}

// ---------------------------------------------------------------------------
// Edge scatter-add: one wave per edge; lane handles 4 columns (float4 read).
// ---------------------------------------------------------------------------
__global__ __launch_bounds__(256) void scatter_kernel(
    const float* __restrict__ m, const int* __restrict__ ei,
    float* __restrict__ aggr, int n_edges) {
    const int t = blockIdx.x * blockDim.x + threadIdx.x;
    const int e = t >> 5;
    const int lane = t & 31;
    if (e >= n_edges) return;
    const int s = ei[e];
    const int d = ei[n_edges + e];
    const float4 v = *(const float4*)(m + (size_t)s * H + lane * 4);
    float* base = aggr + (size_t)d * H + lane * 4;
    atomicAdd(base + 0, v.x);
    atomicAdd(base + 1, v.y);
    atomicAdd(base + 2, v.z);
    atomicAdd(base + 3, v.w);
}

__global__ void zero_kernel(float* __restrict__ p, int n) {
    int i = blockIdx.x * blockDim.x + threadIdx.x;
    if (i < n) p[i] = 0.0f;
}

// Mean-pool accumulation: one wave per node.
__global__ __launch_bounds__(256) void pool_kernel(
    const float* __restrict__ local, const int* __restrict__ batch,
    float* __restrict__ sums, float* __restrict__ counts, int n_nodes) {
    const int t = blockIdx.x * blockDim.x + threadIdx.x;
    const int node = t >> 5;
    const int lane = t & 31;
    if (node >= n_nodes) return;
    const int g = batch[node];
    const float4 v = *(const float4*)(local + (size_t)node * H + lane * 4);
    float* base = sums + (size_t)g * H + lane * 4;
    atomicAdd(base + 0, v.x);
    atomicAdd(base + 1, v.y);
    atomicAdd(base + 2, v.z);
    atomicAdd(base + 3, v.w);
    if (lane == 0) atomicAdd(&counts[g], 1.0f);
}

// Head: pooled@global_w^T + b, then log_softmax. One wave per graph.
__global__ __launch_bounds__(32) void head_kernel(
    const float* __restrict__ sums, const float* __restrict__ counts,
    const float* __restrict__ gw, const float* __restrict__ gb,
    float* __restrict__ out) {
    const int g = blockIdx.x;
    const int lane = threadIdx.x;
    const float inv = 1.0f / fmaxf(counts[g], 1.0f);
    float d = -1e30f;
    if (lane < NCLASSES) {
        float acc = gb[lane];
        for (int k = 0; k < H; ++k)
            acc += (sums[(size_t)g * H + k] * inv) * gw[(size_t)lane * H + k];
        d = acc;
    }
    float mx = -INFINITY;
    for (int i = 0; i < NCLASSES; ++i) mx = fmaxf(mx, __shfl(d, i, 32));
    float se = 0.0f;
    for (int i = 0; i < NCLASSES; ++i) se += expf(__shfl(d, i, 32) - mx);
    const float lse = mx + logf(se);
    if (lane < NCLASSES) out[(size_t)g * NCLASSES + lane] = d - lse;
}

// ---------------------------------------------------------------------------
extern "C" void kernel_launch(void* const* d_in, const int* in_sizes, int n_in,
                              void* d_out, int out_size, void* d_ws, size_t ws_size,
                              hipStream_t stream) {
    const float* x    = (const float*)d_in[0];
    const int*   ei   = (const int*)d_in[1];
    const int*   batch= (const int*)d_in[2];
    const float* W    = (const float*)d_in[3];
    const float* w_ih = (const float*)d_in[4];
    const float* w_hh = (const float*)d_in[5];
    const float* b_ih = (const float*)d_in[6];
    const float* b_hh = (const float*)d_in[7];
    const float* lw   = (const float*)d_in[8];
    const float* lb   = (const float*)d_in[9];
    const float* gw   = (const float*)d_in[10];
    const float* gb   = (const float*)d_in[11];
    float* out = (float*)d_out;

    const int n_nodes  = in_sizes[0] / H;        // 50000
    const int n_edges  = in_sizes[1] / 2;        // 800000
    const int n_layers = in_sizes[3] / (H * H);  // 4

    const size_t hN = (size_t)n_nodes * H;
    float* h      = (float*)d_ws;
    float* m      = h + hN;
    float* aggr   = m + hN;
    float* sums   = aggr + hN;
    float* counts = sums + (size_t)NGRAPHS * H;

    // h = x  (H already equals hidden dim -> no padding needed)
    hipMemcpyAsync(h, x, hN * sizeof(float), hipMemcpyDeviceToDevice, stream);

    const int mtiles = (n_nodes + 15) / 16;                 // 3125
    const int gemmH_blocks = (mtiles * (H / 16) + 7) / 8;   // 8 waves/block
    const int zeroN_blocks = ((int)hN + 255) / 256;
    const int scat_blocks  = (n_edges * 32 + 255) / 256;

    for (int l = 0; l < n_layers; ++l) {
        // m = h @ W[l]
        gemm_kernel<false, false, false><<<gemmH_blocks, 256, 0, stream>>>(
            h, W + (size_t)l * H * H, nullptr, m, n_nodes, H, H);
        // aggr = 0
        zero_kernel<<<zeroN_blocks, 256, 0, stream>>>(aggr, (int)hN);
        // aggr[dst] += m[src]
        scatter_kernel<<<scat_blocks, 256, 0, stream>>>(m, ei, aggr, n_edges);
        // h = GRUCell(aggr, h)   (in place)
        gru_kernel<<<mtiles, 256, 0, stream>>>(aggr, h, w_ih, w_hh, b_ih, b_hh);
    }

    // local = relu(h @ local_w^T + local_b)   (stored in m)
    gemm_kernel<true, true, true><<<gemmH_blocks, 256, 0, stream>>>(
        h, lw, lb, m, n_nodes, H, H);

    // mean pool
    zero_kernel<<<(NGRAPHS * H + NGRAPHS + 255) / 256, 256, 0, stream>>>(
        sums, NGRAPHS * H + NGRAPHS);  // sums and counts are contiguous
    pool_kernel<<<(n_nodes * 32 + 255) / 256, 256, 0, stream>>>(
        m, batch, sums, counts, n_nodes);

    // classifier head + log_softmax
    head_kernel<<<NGRAPHS, 32, 0, stream>>>(sums, counts, gw, gb, out);
}